// EfficientReynoldsFeatureOperator_28338194219464
// MI455X (gfx1250) — compile-verified
//
#include <hip/hip_runtime.h>

// EfficientReynoldsFeatureOperator for MI455X (gfx1250), compile-only tuned.
//
// Math identity: mean_n(x @ W) * pw  ==  ((mean_n x) @ W) * pw broadcast over n.
//   x: (8, 8192, 512) f32, W: (512,128) f32, pw: (128,) f32 -> out (8,8192,128) f32
//
// Roofline: ~128 MiB read (x) + ~32 MiB write (out) at 23.3 TB/s => ~7 us; purely
// bandwidth bound. The 8x512x128 projection is exact-precision fp32 WMMA
// (V_WMMA_F32_16X16X4_F32), with the 16x512 A matrix zero-padded in workspace so
// the fragment loads are branch-free (EXEC stays all-1s, no saveexec churn).
//
// Pipeline (deterministic, no atomics):
//   K1 reduce_seq : partials[b][chunk][d] = sum over 128 rows   (streams 128 MiB)
//   K2 mean_pad   : meanp[m][d], m<8 = (1/8192)*sum_c partials, m>=8 = 0
//   K3 proj_wmma  : proj[b][r] = meanp @ W * pw  via v_wmma_f32_16x16x4_f32
//   K4 broadcast  : out[b][n][r] = proj[b][r]    (32 MiB float4 stores)

typedef float v2f __attribute__((ext_vector_type(2)));
typedef float v8f __attribute__((ext_vector_type(8)));

#define BB     8
#define MM     16          // WMMA M (b padded 8 -> 16)
#define NN     8192
#define DIN    512
#define DRED   128
#define CHUNKS 64
#define ROWS   (NN / CHUNKS)   // 128 rows per chunk

// ---------------- K1: streaming column reduction over the seq axis ----------
__global__ void __launch_bounds__(DIN)
reduce_seq_kernel(const float* __restrict__ x, float* __restrict__ partials) {
    const int d = threadIdx.x;          // 0..511  (contiguous -> coalesced)
    const int b = blockIdx.x;           // 0..7
    const int c = blockIdx.y;           // 0..63
    const long base = ((long)(b * NN + c * ROWS)) * DIN + d;

    float acc = 0.0f;
    #pragma unroll 4
    for (int i = 0; i < ROWS; ++i) {
        // gfx1250 speculative prefetch (global_prefetch_b8) ~48 KiB ahead
        if (i + 24 < ROWS)
            __builtin_prefetch(&x[base + (long)(i + 24) * DIN], 0, 0);
        acc += x[base + (long)i * DIN];
    }
    partials[(b * CHUNKS + c) * DIN + d] = acc;
}

// ---------------- K2: fold chunk partials -> scaled mean, zero-pad to 16 rows
__global__ void __launch_bounds__(DIN)
mean_pad_kernel(const float* __restrict__ partials, float* __restrict__ meanp) {
    const int d = threadIdx.x;
    const int m = blockIdx.x;           // 0..15
    float acc = 0.0f;
    if (m < BB) {
        #pragma unroll 8
        for (int c = 0; c < CHUNKS; ++c)
            acc += partials[(m * CHUNKS + c) * DIN + d];
        acc *= (1.0f / (float)NN);
    }
    meanp[m * DIN + d] = acc;           // rows 8..15 written as 0.0f
}

// ---------------- K3: proj = meanp(16x512) @ W(512x128) via fp32 WMMA -------
// 8 waves; wave w owns the N-tile r = [16w, 16w+16). K loop: 512/4 = 128 WMMAs.
// A 16x4 f32 layout (ISA 7.12.2): lanes 0-15 -> K={k0,k0+1}, lanes 16-31 -> K={k0+2,k0+3}.
// B 4x16: lane%16 = N column, same K split by lane-half.
// C/D: VGPR j = row M=j for lanes 0-15 (M=j+8 in lanes 16-31 = zero padding).
__global__ void __launch_bounds__(256)
proj_wmma_kernel(const float* __restrict__ meanp,
                 const float* __restrict__ W,
                 const float* __restrict__ pw,
                 float* __restrict__ projw) {
    const int wave = threadIdx.x >> 5;   // 0..7 -> r tile
    const int lane = threadIdx.x & 31;
    const int lr   = lane & 15;
    const int half = lane >> 4;
    const int rb   = wave * 16;

    const float* __restrict__ arow = meanp + lr * DIN;    // A row M=lr (zero-padded)

    v8f c = {};
    for (int k0 = 0; k0 < DIN; k0 += 4) {
        const int k = k0 + 2 * half;

        v2f a;                                  // A[lr][k], A[lr][k+1] -- branch-free
        a.x = arow[k];
        a.y = arow[k + 1];

        v2f bv;                                 // B[k][rb+lr], B[k+1][rb+lr] (W row-major)
        bv.x = W[(long)k       * DRED + rb + lr];
        bv.y = W[(long)(k + 1) * DRED + rb + lr];

        c = __builtin_amdgcn_wmma_f32_16x16x4_f32(
                /*neg_a=*/false, a, /*neg_b=*/false, bv,
                /*c_mod=*/(short)0, c, /*reuse_a=*/false, /*reuse_b=*/false);
    }

    // rows M=0..7 live in VGPR j=0..7, lanes 0-15 (column N = rb+lane)
    if (half == 0) {
        const float p = pw[rb + lr];
        #pragma unroll
        for (int j = 0; j < BB; ++j)
            projw[j * DRED + rb + lr] = c[j] * p;
    }
}

// ---------------- K4: broadcast proj row across the whole seq axis ----------
__global__ void __launch_bounds__(256)
broadcast_kernel(const float* __restrict__ projw, float* __restrict__ out) {
    __shared__ float4 srow[DRED / 4];            // 32 x float4 = one 128-float row
    const int b  = blockIdx.x;
    const int n0 = blockIdx.y * 64;              // 64 seq rows per block

    if (threadIdx.x < DRED / 4) {
        const float4* p4 = reinterpret_cast<const float4*>(projw + b * DRED);
        srow[threadIdx.x] = p4[threadIdx.x];
    }
    __syncthreads();

    float4* out4 = reinterpret_cast<float4*>(out);
    const long rowbase = ((long)b * NN + n0) * (DRED / 4);
    #pragma unroll
    for (int p = 0; p < 8; ++p) {
        const int e   = p * 256 + threadIdx.x;   // 0..2047 float4 slots (64 rows x 32)
        const int row = e >> 5;
        const int col = e & 31;
        out4[rowbase + (long)row * (DRED / 4) + col] = srow[col];
    }
}

extern "C" void kernel_launch(void* const* d_in, const int* in_sizes, int n_in,
                              void* d_out, int out_size, void* d_ws, size_t ws_size,
                              hipStream_t stream) {
    const float* x  = (const float*)d_in[0];   // (8, 8192, 512)
    const float* W  = (const float*)d_in[1];   // (512, 128)
    const float* pw = (const float*)d_in[2];   // (128,)
    float* out = (float*)d_out;                // (8, 8192, 128)

    float* partials = (float*)d_ws;                      // 8*64*512 f32 = 1 MiB
    float* meanp    = partials + BB * CHUNKS * DIN;      // 16*512    = 32 KiB (zero-padded)
    float* projw    = meanp + MM * DIN;                  // 8*128     = 4 KiB

    reduce_seq_kernel<<<dim3(BB, CHUNKS), DIN, 0, stream>>>(x, partials);
    mean_pad_kernel <<<dim3(MM),          DIN, 0, stream>>>(partials, meanp);
    proj_wmma_kernel<<<1,                 256, 0, stream>>>(meanp, W, pw, projw);
    broadcast_kernel<<<dim3(BB, NN / 64), 256, 0, stream>>>(projw, out);
}